// PagedAttention_8727373545618
// MI455X (gfx1250) — compile-verified
//
#include <hip/hip_runtime.h>
#include <hip/hip_bf16.h>

typedef float v2f __attribute__((ext_vector_type(2)));
typedef float v8f __attribute__((ext_vector_type(8)));
typedef unsigned v4u __attribute__((ext_vector_type(4)));
typedef int v4i __attribute__((ext_vector_type(4)));
typedef int v8i __attribute__((ext_vector_type(8)));

#define NW 4
#define NEG_INF_F (-1e30f)

__device__ __forceinline__ float rmax16(float x) {
  x = fmaxf(x, __shfl_xor(x, 8, 32));
  x = fmaxf(x, __shfl_xor(x, 4, 32));
  x = fmaxf(x, __shfl_xor(x, 2, 32));
  x = fmaxf(x, __shfl_xor(x, 1, 32));
  return x;
}
__device__ __forceinline__ float radd16(float x) {
  x += __shfl_xor(x, 8, 32);
  x += __shfl_xor(x, 4, 32);
  x += __shfl_xor(x, 2, 32);
  x += __shfl_xor(x, 1, 32);
  return x;
}
__device__ __forceinline__ float radd32(float x) {
  x += __shfl_xor(x, 16, 32);
  return radd16(x);
}

// Issue one TDM 2D tile load: 16 rows x 512B, row stride 4KB, into LDS.
// D# group1 is constant for all tiles (built by caller); group0 varies.
__device__ __forceinline__ void tdm_issue(unsigned long long ga, unsigned lds_addr,
                                          v8i g1) {
  v4u g0;
  g0.x = 1u;  // count=1, is_restore=0, gather off
  g0.y = (unsigned)__builtin_amdgcn_readfirstlane((int)lds_addr);
  g0.z = (unsigned)__builtin_amdgcn_readfirstlane((int)(unsigned)ga);
  g0.w = (unsigned)__builtin_amdgcn_readfirstlane(
      (int)((((unsigned)(ga >> 32)) & 0x01FFFFFFu) | (2u << 30)));  // addr hi | type=2
  v4i z4 = {};
  v8i z8 = {};
  __builtin_amdgcn_tensor_load_to_lds(g0, g1, z4, z4, z8, 0);
}

// B=32, H=32, KVH=8, G=4, D=128, BLOCK_SIZE=16, MAX_BLOCKS=128
__global__ __launch_bounds__(128) void paged_attn_kernel(
    const float* __restrict__ q, const float* __restrict__ knew,
    const float* __restrict__ vnew, const float* __restrict__ kc,
    const float* __restrict__ vc, const int* __restrict__ btab,
    const int* __restrict__ ctxl, float* __restrict__ out) {
  const int wg = blockIdx.x;
  const int b = wg >> 3;        // batch
  const int kvh = wg & 7;       // kv head
  const int tid = threadIdx.x;
  const int w = tid >> 5;       // wave id 0..3
  const int lane = tid & 31;
  const int ln = lane & 15;           // row/col within 16
  const int kk = (lane >> 4) << 1;    // K-pair offset (0 or 2)

  __shared__ __align__(16) float lds_q[512];         // 4 heads x 128 (rope'd q)
  __shared__ __align__(16) float lds_k[128];         // rope'd new key
  __shared__ __align__(16) float lds_v[128];         // new value
  __shared__ __align__(16) float lds_p[NW * 256];    // per-wave 16x16 probs tile
  __shared__ __align__(16) float lds_m[NW * 4];
  __shared__ __align__(16) float lds_l[NW * 4];
  __shared__ __align__(16) float lds_acc[NW * 4 * 128];
  __shared__ __align__(16) float kstage[NW][2][2048]; // double-buffered K tiles
  __shared__ __align__(16) float vstage[NW][2][2048]; // double-buffered V tiles

  const int ctx = ctxl[b];
  const int tok = ctx - 1;
  const float pos = (float)ctx;   // reference uses pos = context_lens
  const float scale = 0.08838834764831845f; // 1/sqrt(128)

  // ---- RoPE on query (4 heads of this group) into LDS ----
  for (int idx = tid; idx < 512; idx += 128) {
    int h = idx >> 7, d = idx & 127;
    int i = d & 63;
    float inv = __expf(-((float)i * (1.0f / 64.0f)) * 9.210340371976184f);
    float ang = pos * inv;
    float cs = __cosf(ang), sn = __sinf(ang);
    const float* qrow = q + ((size_t)(b * 32 + kvh * 4 + h)) * 128;
    float a0 = qrow[d];
    float a1 = (d < 64) ? qrow[d + 64] : qrow[d - 64];
    lds_q[h * 128 + d] = (d < 64) ? (a0 * cs - a1 * sn) : (a0 * cs + a1 * sn);
  }
  // ---- RoPE on new key + copy new value ----
  {
    int d = tid, i = d & 63;
    float inv = __expf(-((float)i * (1.0f / 64.0f)) * 9.210340371976184f);
    float ang = pos * inv;
    float cs = __cosf(ang), sn = __sinf(ang);
    const float* krow = knew + ((size_t)(b * 8 + kvh)) * 128;
    float a0 = krow[d];
    float a1 = (d < 64) ? krow[d + 64] : krow[d - 64];
    lds_k[d] = (d < 64) ? (a0 * cs - a1 * sn) : (a0 * cs + a1 * sn);
    lds_v[d] = vnew[((size_t)(b * 8 + kvh)) * 128 + d];
  }
  // zero probs tiles (rows 4..15 stay zero => unused WMMA M rows contribute 0)
  for (int i2 = tid; i2 < NW * 256; i2 += 128) lds_p[i2] = 0.0f;
  __syncthreads();

  const int* btrow = btab + b * 128;
  const int ntiles = (ctx + 15) >> 4; // one tile == one 16-token cache block

  float m[4], l[4];
  #pragma unroll
  for (int h = 0; h < 4; ++h) { m[h] = NEG_INF_F; l[h] = 0.0f; }
  v8f accv[8];
  #pragma unroll
  for (int nc = 0; nc < 8; ++nc) accv[nc] = {};

  const float* qb = lds_q + (lane & 3) * 128 + kk;
  float* ldsp_w = lds_p + w * 256;

  // D# group1: 2D tensor, 4B elems, tensor_dim=[128,16], tile=[128,16], stride0=1024
  const v8i g1 = {0x00020000, (int)(128u << 16), (int)(16u << 16),
                  (int)(128u << 16), 16, 1024, 0, 0};
  const unsigned ka[2] = {(unsigned)(uintptr_t)&kstage[w][0][0],
                          (unsigned)(uintptr_t)&kstage[w][1][0]};
  const unsigned va[2] = {(unsigned)(uintptr_t)&vstage[w][0][0],
                          (unsigned)(uintptr_t)&vstage[w][1][0]};
  const unsigned long long kbase = (unsigned long long)(uintptr_t)kc + (size_t)kvh * 512;
  const unsigned long long vbase = (unsigned long long)(uintptr_t)vc + (size_t)kvh * 512;

  // prologue: start DMA for this wave's first tile
  if (w < ntiles) {
    const unsigned long long off = (unsigned long long)btrow[w] * 65536ull;
    tdm_issue(kbase + off, ka[0], g1);
    tdm_issue(vbase + off, va[0], g1);
  }

  int cur = 0;
  for (int t = w; t < ntiles; t += NW) {
    const int tn = t + NW;
    if (tn < ntiles) {
      const unsigned long long off = (unsigned long long)btrow[tn] * 65536ull;
      tdm_issue(kbase + off, ka[cur ^ 1], g1);
      tdm_issue(vbase + off, va[cur ^ 1], g1);
      __builtin_amdgcn_s_wait_tensorcnt(2); // older K/V pair has landed
    } else {
      __builtin_amdgcn_s_wait_tensorcnt(0);
    }

    // ---- scores: q (16x4 per step) x K^T (4x16 per step) from staged LDS ----
    const float* kb = &kstage[w][cur][0] + ln * 128 + kk;
    v8f c = {};
    #pragma unroll
    for (int d0 = 0; d0 < 128; d0 += 4) {
      v2f a = *(const v2f*)(qb + d0);
      v2f bv = *(const v2f*)(kb + d0);
      c = __builtin_amdgcn_wmma_f32_16x16x4_f32(false, a, false, bv, (short)0, c,
                                                false, false);
    }
    // ---- mask + online softmax ----
    const int s = (t << 4) + ln;
    const bool valid = (s < ctx) && (s != tok);
    float fac[4], p[4];
    #pragma unroll
    for (int h = 0; h < 4; ++h) {
      float x = valid ? c[h] * scale : NEG_INF_F;
      float tm = rmax16(x);
      float mn = fmaxf(m[h], tm);
      fac[h] = __expf(m[h] - mn);
      p[h] = valid ? __expf(x - mn) : 0.0f;
      l[h] = l[h] * fac[h] + radd16(p[h]);
      m[h] = mn;
    }
    #pragma unroll
    for (int nc = 0; nc < 8; ++nc) {
      #pragma unroll
      for (int h = 0; h < 4; ++h) accv[nc][h] *= fac[h];
    }
    if (lane < 16) {
      #pragma unroll
      for (int h = 0; h < 4; ++h) ldsp_w[h * 16 + lane] = p[h];
    }
    // ---- out += P (16x16) x V (16x128) from staged LDS ----
    const float* vb = &vstage[w][cur][0];
    #pragma unroll
    for (int k0 = 0; k0 < 16; k0 += 4) {
      v2f pa = *(const v2f*)(ldsp_w + ln * 16 + k0 + kk);
      const float* vrow = vb + (k0 + kk) * 128 + ln;
      #pragma unroll
      for (int nc = 0; nc < 8; ++nc) {
        v2f bv;
        bv.x = vrow[nc * 16];
        bv.y = vrow[128 + nc * 16];
        accv[nc] = __builtin_amdgcn_wmma_f32_16x16x4_f32(false, pa, false, bv,
                                                         (short)0, accv[nc],
                                                         false, false);
      }
    }
    cur ^= 1;
  }

  // ---- fold in the freshly inserted (k_rot, v_new) token (wave 0 only) ----
  if (w == 0) {
    float part[4] = {0.f, 0.f, 0.f, 0.f};
    #pragma unroll
    for (int d0 = 0; d0 < 128; d0 += 32) {
      float kvv = lds_k[d0 + lane];
      #pragma unroll
      for (int h = 0; h < 4; ++h) part[h] += lds_q[h * 128 + d0 + lane] * kvv;
    }
    float fac[4], p[4];
    #pragma unroll
    for (int h = 0; h < 4; ++h) {
      float sh = radd32(part[h]) * scale;
      float mn = fmaxf(m[h], sh);
      fac[h] = __expf(m[h] - mn);
      p[h] = __expf(sh - mn);
      l[h] = l[h] * fac[h] + p[h];
      m[h] = mn;
    }
    #pragma unroll
    for (int nc = 0; nc < 8; ++nc) {
      float vv = lds_v[nc * 16 + ln];
      #pragma unroll
      for (int h = 0; h < 4; ++h) accv[nc][h] = accv[nc][h] * fac[h] + p[h] * vv;
    }
  }

  // ---- cross-wave merge ----
  if (lane == 0) {
    #pragma unroll
    for (int h = 0; h < 4; ++h) lds_m[w * 4 + h] = m[h];
  }
  __syncthreads();
  float Mh[4];
  #pragma unroll
  for (int h = 0; h < 4; ++h) {
    float mm = NEG_INF_F;
    for (int wv = 0; wv < NW; ++wv) mm = fmaxf(mm, lds_m[wv * 4 + h]);
    Mh[h] = mm;
  }
  float g[4];
  #pragma unroll
  for (int h = 0; h < 4; ++h) g[h] = __expf(m[h] - Mh[h]);
  if (lane == 0) {
    #pragma unroll
    for (int h = 0; h < 4; ++h) lds_l[w * 4 + h] = l[h] * g[h];
  }
  if (lane < 16) {
    float* dst = lds_acc + (size_t)(w * 4) * 128;
    #pragma unroll
    for (int nc = 0; nc < 8; ++nc) {
      #pragma unroll
      for (int h = 0; h < 4; ++h)
        dst[h * 128 + nc * 16 + lane] = accv[nc][h] * g[h];
    }
  }
  __syncthreads();

  // ---- final: sum over waves, divide, store ----
  for (int idx = tid; idx < 512; idx += 128) {
    int h = idx >> 7, d = idx & 127;
    float o = 0.0f, L = 0.0f;
    for (int wv = 0; wv < NW; ++wv) {
      o += lds_acc[(wv * 4 + h) * 128 + d];
      L += lds_l[wv * 4 + h];
    }
    out[((size_t)(b * 32 + kvh * 4 + h)) * 128 + d] = o / L;
  }
}

extern "C" void kernel_launch(void* const* d_in, const int* in_sizes, int n_in,
                              void* d_out, int out_size, void* d_ws, size_t ws_size,
                              hipStream_t stream) {
  (void)in_sizes; (void)n_in; (void)out_size; (void)d_ws; (void)ws_size;
  const float* q  = (const float*)d_in[0];
  const float* k  = (const float*)d_in[1];
  const float* v  = (const float*)d_in[2];
  const float* kc = (const float*)d_in[3];
  const float* vc = (const float*)d_in[4];
  const int* bt   = (const int*)d_in[5];
  const int* cl   = (const int*)d_in[6];
  float* out = (float*)d_out;
  dim3 grid(256), block(128);
  hipLaunchKernelGGL(paged_attn_kernel, grid, block, 0, stream,
                     q, k, v, kc, vc, bt, cl, out);
}